// NonParaGCNConv_34668976013866
// MI455X (gfx1250) — compile-verified
//
#include <hip/hip_runtime.h>

#define EPS_SCALE 0.5f
#define FDIM 64

// ---------------------------------------------------------------------------
// Pass 1: out-degree of each node (int atomics, resolved at L2)
// ---------------------------------------------------------------------------
__global__ void gcn_deg_kernel(const int* __restrict__ src,
                               int* __restrict__ deg, int E) {
    int i = blockIdx.x * blockDim.x + threadIdx.x;
    if (i < E) {
        atomicAdd(&deg[src[i]], 1);
    }
}

// ---------------------------------------------------------------------------
// Pass 2: norm[i] = rsqrt(max(deg, 1))
// ---------------------------------------------------------------------------
__global__ void gcn_norm_kernel(const int* __restrict__ deg,
                                float* __restrict__ norm, int N) {
    int i = blockIdx.x * blockDim.x + threadIdx.x;
    if (i < N) {
        float d = (float)deg[i];
        norm[i] = rsqrtf(fmaxf(d, 1.0f));
    }
}

// ---------------------------------------------------------------------------
// Pass 3: out = EPS * x  (streaming float4; also deterministically
// re-initializes d_out each call, overwriting harness poison)
// ---------------------------------------------------------------------------
__global__ void gcn_resid_kernel(const float4* __restrict__ x4,
                                 float4* __restrict__ out4, int n4) {
    int i = blockIdx.x * blockDim.x + threadIdx.x;
    if (i < n4) {
        float4 v = x4[i];
        v.x *= EPS_SCALE; v.y *= EPS_SCALE; v.z *= EPS_SCALE; v.w *= EPS_SCALE;
        out4[i] = v;
    }
}

// ---------------------------------------------------------------------------
// Pass 4: one wave32 per edge.
//   lane L handles features [2L, 2L+1]  -> 32 lanes x float2 = 64 floats,
//   i.e. one coalesced 256B row load of x[src] and 64 f32 atomics into
//   out[dst]. Indices + norms are wave-uniform -> scalar loads.
// ---------------------------------------------------------------------------
__global__ void gcn_edge_kernel(const float* __restrict__ x,
                                const int* __restrict__ src,
                                const int* __restrict__ dst,
                                const float* __restrict__ norm,
                                float* __restrict__ out, int E) {
    int flat = blockIdx.x * blockDim.x + threadIdx.x;
    int wave = flat >> 5;          // wave32: one edge per wave
    int lane = threadIdx.x & 31;
    if (wave >= E) return;

    // Force the edge id onto the scalar path: src/dst/norm fetches become
    // s_load's (KMcnt), leaving the vector pipe for the row gather/scatter.
    int we = __builtin_amdgcn_readfirstlane(wave);
    int s = src[we];
    int d = dst[we];
    float w = norm[s] * norm[d];

    // Prefetch a future gather row (lowers to global_prefetch_b8 on gfx1250)
    // to hide the latency of rows that miss in the 192MB L2.
    int pe = we + 16;
    if (pe >= E) pe = we;
    int ps = src[pe];
    __builtin_prefetch(&x[(size_t)ps * FDIM + lane * 2], 0, 1);

    const float2* xr = (const float2*)(x + (size_t)s * FDIM);
    float2 v = xr[lane];

    float* orow = out + (size_t)d * FDIM + lane * 2;
    atomicAdd(orow,     v.x * w);   // global_atomic_add_f32 (no return)
    atomicAdd(orow + 1, v.y * w);
}

// ---------------------------------------------------------------------------
// Launcher
// ---------------------------------------------------------------------------
extern "C" void kernel_launch(void* const* d_in, const int* in_sizes, int n_in,
                              void* d_out, int out_size, void* d_ws, size_t ws_size,
                              hipStream_t stream) {
    const float* x  = (const float*)d_in[0];
    const int* src  = (const int*)d_in[1];
    const int* dst  = (const int*)d_in[2];
    float* out      = (float*)d_out;

    const int N = in_sizes[0] / FDIM;
    const int E = in_sizes[1];

    // Workspace layout: [ deg : N ints | norm : N floats ]
    int*   deg  = (int*)d_ws;
    float* norm = (float*)((char*)d_ws + (size_t)N * sizeof(int));

    hipMemsetAsync(deg, 0, (size_t)N * sizeof(int), stream);

    gcn_deg_kernel<<<(E + 255) / 256, 256, 0, stream>>>(src, deg, E);

    gcn_norm_kernel<<<(N + 255) / 256, 256, 0, stream>>>(deg, norm, N);

    const int n4 = (N * FDIM) / 4;
    gcn_resid_kernel<<<(n4 + 255) / 256, 256, 0, stream>>>(
        (const float4*)x, (float4*)out, n4);

    // one wave32 (32 threads) per edge, 8 waves per 256-thread block
    long edge_threads = (long)E * 32;
    int  edge_blocks  = (int)((edge_threads + 255) / 256);
    gcn_edge_kernel<<<edge_blocks, 256, 0, stream>>>(x, src, dst, norm, out, E);
}